// ChebConv_48473000903659
// MI455X (gfx1250) — compile-verified
//
#include <hip/hip_runtime.h>

typedef float v2f __attribute__((ext_vector_type(2)));
typedef float v8f __attribute__((ext_vector_type(8)));

#define D 128

// ---------------- utility ----------------
__global__ void zero_kernel(float* __restrict__ p, long n) {
  long i = (long)blockIdx.x * blockDim.x + threadIdx.x;
  if (i < n) p[i] = 0.0f;
}

__global__ void deg_kernel(const int* __restrict__ dst, float* __restrict__ deg, int n_edges) {
  int e = blockIdx.x * blockDim.x + threadIdx.x;
  if (e < n_edges) unsafeAtomicAdd(&deg[dst[e]], 1.0f);
}

__global__ void norm_kernel(float* __restrict__ norm, int n) {
  int i = blockIdx.x * blockDim.x + threadIdx.x;
  if (i < n) {
    float d = norm[i];
    norm[i] = 1.0f / sqrtf(fmaxf(d, 1.0f));
  }
}

// ---------------- SpMM scatter: h[dst] += x[src] * norm[src] ----------------
// one wave (32 lanes) per edge, float4 per lane -> 128 floats per edge
__global__ __launch_bounds__(256) void spmm_kernel(const float* __restrict__ x,
                                                   const float* __restrict__ norm,
                                                   const int* __restrict__ src,
                                                   const int* __restrict__ dst,
                                                   float* __restrict__ h, int n_edges) {
  int wave = blockIdx.x * (blockDim.x >> 5) + (threadIdx.x >> 5);
  int lane = threadIdx.x & 31;
  if (wave >= n_edges) return;
  int s = src[wave];
  int d = dst[wave];
  float ns = norm[s];
  const float4* xs = (const float4*)(x + (size_t)s * D);
  float4 v = xs[lane];
  float* hd = h + (size_t)d * D + lane * 4;
  unsafeAtomicAdd(hd + 0, v.x * ns);
  unsafeAtomicAdd(hd + 1, v.y * ns);
  unsafeAtomicAdd(hd + 2, v.z * ns);
  unsafeAtomicAdd(hd + 3, v.w * ns);
}

// ---------------- Chebyshev recurrence ----------------
// Tx1 = -(2/lam)*(norm*h) + Tx0*(2/lam - 1)
__global__ void cheb1_kernel(const float* __restrict__ feat, const float* __restrict__ h,
                             const float* __restrict__ norm, const float* __restrict__ lam_p,
                             float* __restrict__ tx1, long n) {
  long i = (long)blockIdx.x * blockDim.x + threadIdx.x;
  if (i >= n) return;
  int node = (int)(i >> 7);
  float c = 2.0f / lam_p[0];
  tx1[i] = -c * (norm[node] * h[i]) + feat[i] * (c - 1.0f);
}

// Tx2 = -(4/lam)*(norm*h2) + Tx1*(4/lam - 2) - Tx0
__global__ void cheb2_kernel(const float* __restrict__ feat, const float* __restrict__ tx1,
                             const float* __restrict__ h, const float* __restrict__ norm,
                             const float* __restrict__ lam_p, float* __restrict__ tx2, long n) {
  long i = (long)blockIdx.x * blockDim.x + threadIdx.x;
  if (i >= n) return;
  int node = (int)(i >> 7);
  float c = 4.0f / lam_p[0];
  tx2[i] = -c * (norm[node] * h[i]) + tx1[i] * (c - 2.0f) - feat[i];
}

// ---------------- fused WMMA GEMM: out = Tx0 W0^T + Tx1 W1^T + Tx2 W2^T + bias --------
// one wave -> one 16x16 output tile; 8 waves per block cover all 128 output cols
__global__ __launch_bounds__(256) void gemm_kernel(const float* __restrict__ tx0,
                                                   const float* __restrict__ tx1,
                                                   const float* __restrict__ tx2,
                                                   const float* __restrict__ W,
                                                   const float* __restrict__ bias,
                                                   float* __restrict__ out) {
  const int lane  = threadIdx.x & 31;
  const int waveN = threadIdx.x >> 5;        // N tile index, 0..7
  const int mbase = blockIdx.x * 16;
  const int nbase = waveN * 16;
  const int mn    = lane & 15;               // A row / B col within tile
  const int klo   = (lane >> 4) << 1;        // 0 or 2 (K split at lane 16)

  const float* Xs[3] = {tx0, tx1, tx2};
  v8f acc = {};
#pragma unroll
  for (int c = 0; c < 3; ++c) {
    // A: X[mbase+mn][k+klo .. +1]   B: W_c^T -> load W_c[nbase+mn][k+klo .. +1]
    const float2* ap = (const float2*)(Xs[c] + (size_t)(mbase + mn) * D + klo);
    const float2* bp = (const float2*)(W + (size_t)c * D * D + (size_t)(nbase + mn) * D + klo);
#pragma unroll
    for (int k = 0; k < D / 4; ++k) {
      float2 af = ap[2 * k];
      float2 bf = bp[2 * k];
      v2f a; a.x = af.x; a.y = af.y;
      v2f b; b.x = bf.x; b.y = bf.y;
      acc = __builtin_amdgcn_wmma_f32_16x16x4_f32(false, a, false, b, (short)0, acc,
                                                  false, false);
    }
  }
  float bv = bias[nbase + mn];
  const int mtop = (lane >> 4) << 3;         // rows v (lanes 0-15) / v+8 (lanes 16-31)
#pragma unroll
  for (int v = 0; v < 8; ++v) {
    out[(size_t)(mbase + mtop + v) * D + nbase + mn] = acc[v] + bv;
  }
}

// ---------------- launch ----------------
extern "C" void kernel_launch(void* const* d_in, const int* in_sizes, int n_in,
                              void* d_out, int out_size, void* d_ws, size_t ws_size,
                              hipStream_t stream) {
  const float* feat = (const float*)d_in[0];
  const float* W    = (const float*)d_in[1];
  const float* bias = (const float*)d_in[2];
  const float* lam  = (const float*)d_in[3];
  const int*   src  = (const int*)d_in[4];
  const int*   dst  = (const int*)d_in[5];
  float* out = (float*)d_out;

  const int  n_nodes = in_sizes[0] / D;
  const int  n_edges = in_sizes[4];
  const long nd      = (long)n_nodes * D;

  float* ws   = (float*)d_ws;
  long   npad = ((long)n_nodes + 255) & ~255L;
  float* norm = ws;
  float* h    = ws + npad;
  float* tx1  = h + nd;
  float* tx2  = tx1 + nd;

  const int T = 256;
  // degree -> norm
  zero_kernel<<<(int)((n_nodes + T - 1) / T), T, 0, stream>>>(norm, n_nodes);
  deg_kernel<<<(n_edges + T - 1) / T, T, 0, stream>>>(dst, norm, n_edges);
  norm_kernel<<<(n_nodes + T - 1) / T, T, 0, stream>>>(norm, n_nodes);
  // h = scatter(feat * norm); Tx1
  zero_kernel<<<(int)((nd + T - 1) / T), T, 0, stream>>>(h, nd);
  spmm_kernel<<<(n_edges + 7) / 8, T, 0, stream>>>(feat, norm, src, dst, h, n_edges);
  cheb1_kernel<<<(int)((nd + T - 1) / T), T, 0, stream>>>(feat, h, norm, lam, tx1, nd);
  // h = scatter(Tx1 * norm); Tx2
  zero_kernel<<<(int)((nd + T - 1) / T), T, 0, stream>>>(h, nd);
  spmm_kernel<<<(n_edges + 7) / 8, T, 0, stream>>>(tx1, norm, src, dst, h, n_edges);
  cheb2_kernel<<<(int)((nd + T - 1) / T), T, 0, stream>>>(feat, tx1, h, norm, lam, tx2, nd);
  // fused 3-term GEMM + bias via f32 WMMA
  gemm_kernel<<<n_nodes / 16, T, 0, stream>>>(feat, tx1, tx2, W, bias, out);
}